// Standardization_57956288692228
// MI455X (gfx1250) — compile-verified
//
#include <hip/hip_runtime.h>
#include <hip/hip_bf16.h>
#include <math.h>

#define EPS 1e-7f

typedef __attribute__((ext_vector_type(4))) float f32x4;

// Each wave32 handles a tile of 64 segments (64 * 10 floats = 2560 bytes).
// Lane L owns segments 2L and 2L+1 (one 80-byte row, 16B aligned).
// Tile is staged global -> LDS with CDNA5 async-copy (ASYNCcnt-tracked),
// which makes the loads perfectly coalesced 128b streams; output is written
// with aligned non-temporal 128b stores.
__global__ __launch_bounds__(320) void standardize_seg10_kernel(
    const float* __restrict__ in, float* __restrict__ out, int numSegs) {
  constexpr int SEG = 10;                       // floats per segment
  constexpr int WAVE_SEGS = 64;                 // segments per wave tile
  constexpr int WAVES_PER_BLOCK = 10;           // blockDim = 320
  __shared__ __align__(16) float tile[WAVES_PER_BLOCK * WAVE_SEGS * SEG];

  const int lane = threadIdx.x & 31;
  const int wave = threadIdx.x >> 5;
  const long long gw = (long long)blockIdx.x * WAVES_PER_BLOCK + wave;
  const long long segBase = gw * WAVE_SEGS;

  float* slice = &tile[wave * WAVE_SEGS * SEG];

  if (segBase + WAVE_SEGS <= (long long)numSegs) {
    // ---------------- fast path: async-copy 2560B tile into LDS ----------
    const char* gsrc = (const char*)in + segBase * (SEG * 4);
    unsigned long long ga = (unsigned long long)gsrc + (unsigned)lane * 16u;
    // Raw 32-bit LDS byte address for the async-copy VDST operand.
    unsigned lds_addr =
        (unsigned)(unsigned long long)(__attribute__((address_space(3))) char*)
            (slice) + (unsigned)lane * 16u;

    // 5 x b128 per lane: each instruction moves a coalesced 512B chunk for
    // the wave. Per ISA, the instruction offset is added to BOTH the global
    // and the LDS address, so one (lds,global) base pair serves all five.
    asm volatile("global_load_async_to_lds_b128 %0, %1, off"
                 :: "v"(lds_addr), "v"(ga) : "memory");
    asm volatile("global_load_async_to_lds_b128 %0, %1, off offset:512"
                 :: "v"(lds_addr), "v"(ga) : "memory");
    asm volatile("global_load_async_to_lds_b128 %0, %1, off offset:1024"
                 :: "v"(lds_addr), "v"(ga) : "memory");
    asm volatile("global_load_async_to_lds_b128 %0, %1, off offset:1536"
                 :: "v"(lds_addr), "v"(ga) : "memory");
    asm volatile("global_load_async_to_lds_b128 %0, %1, off offset:2048"
                 :: "v"(lds_addr), "v"(ga) : "memory");
    // Wave-private tile: only this wave reads what it loaded -> no barrier,
    // just drain the async counter.
    asm volatile("s_wait_asynccnt 0x0" ::: "memory");

    // ---------------- compute: 2 segments (20 floats) per lane ------------
    const float* my = slice + lane * (2 * SEG);
    float v[2 * SEG];
#pragma unroll
    for (int i = 0; i < 2 * SEG; ++i) v[i] = my[i];

    f32x4 o[5];
    float* ov = (float*)o;
#pragma unroll
    for (int s = 0; s < 2; ++s) {
      float sum = 0.0f;
#pragma unroll
      for (int i = 0; i < SEG; ++i) sum += v[s * SEG + i];
      const float m = sum * 0.1f;
      float var = 0.0f;
#pragma unroll
      for (int i = 0; i < SEG; ++i) {
        const float d = v[s * SEG + i] - m;
        var += d * d;
      }
      var *= 0.1f;
      const float inv = 1.0f / (sqrtf(var) + EPS);
#pragma unroll
      for (int i = 0; i < SEG; ++i) ov[s * SEG + i] = (v[s * SEG + i] - m) * inv;
    }

    // ---------------- aligned 128b non-temporal stores --------------------
    f32x4* og = (f32x4*)((char*)out + segBase * (SEG * 4) + (long long)lane * 80);
#pragma unroll
    for (int k = 0; k < 5; ++k) __builtin_nontemporal_store(o[k], og + k);
  } else {
    // ---------------- tail path (not taken for the 40M-element shape) -----
    for (long long s = segBase + lane; s < segBase + WAVE_SEGS; s += 32) {
      if (s >= (long long)numSegs) break;
      const float* p = in + s * SEG;
      float v[SEG];
      float sum = 0.0f;
#pragma unroll
      for (int i = 0; i < SEG; ++i) { v[i] = p[i]; sum += v[i]; }
      const float m = sum * 0.1f;
      float var = 0.0f;
#pragma unroll
      for (int i = 0; i < SEG; ++i) { const float d = v[i] - m; var += d * d; }
      var *= 0.1f;
      const float inv = 1.0f / (sqrtf(var) + EPS);
      float* q = out + s * SEG;
#pragma unroll
      for (int i = 0; i < SEG; ++i) q[i] = (v[i] - m) * inv;
    }
  }
}

extern "C" void kernel_launch(void* const* d_in, const int* in_sizes, int n_in,
                              void* d_out, int out_size, void* d_ws, size_t ws_size,
                              hipStream_t stream) {
  const float* x = (const float*)d_in[0];
  float* out = (float*)d_out;
  const int n = in_sizes[0];          // 40,000,000 floats
  const int numSegs = n / 10;         // 4,000,000 segments
  const int wavesNeeded = (numSegs + 63) / 64;          // 62,500
  const int blocks = (wavesNeeded + 9) / 10;            // 6,250
  standardize_seg10_kernel<<<blocks, 320, 0, stream>>>(x, out, numSegs);
}